// GRAPH_MAMBA_29162827939963
// MI455X (gfx1250) — compile-verified
//
#include <hip/hip_runtime.h>
#include <hip/hip_bf16.h>

// Problem dims (from reference)
#define BB 16
#define LL 1024
#define DD 64
#define DI 128
#define SS 16
#define RR 4
#define KK 4
#define HH 256
#define NN 1024
#define EDIM 10
#define PLDIM 96

typedef float v2f __attribute__((ext_vector_type(2)));
typedef float v8f __attribute__((ext_vector_type(8)));

// ---------------- workspace layout (floats) ----------------
#define O_X      0u            // (B,L,D)   persistent x
#define O_YF     1048576u      // (B,L,D)
#define O_YB     2097152u      // (B,L,D)   (flipped time order)
#define O_XFLIP  3145728u      // (B,L,D)
#define O_XZ     4194304u      // (B,L,2DI) 4.19M
#define O_XC     8388608u      // (B,L,DI)  2.10M
#define O_DBC    10485760u     // (B,L,36)
#define O_YBUF   11075584u     // (B,L,DI)
// graph-phase aliases (mamba scratch no longer live)
#define O_SUP    O_XFLIP
#define O_CHEB2  O_XZ
#define O_XG1    (O_XZ + 1048576u)
#define O_XG2    (O_XZ + 2097152u)
#define O_OUTPRE O_XC

#define WMMA_F32(a, b, c) \
  __builtin_amdgcn_wmma_f32_16x16x4_f32(false, (a), false, (b), (short)0, (c), false, false)

// =====================================================================
// f32 WMMA GEMM, 4x2 register-blocked: one wave computes a 64x32 C tile.
// Requires K % 4 == 0, lda even, M % 16 == 0 recommended.
// Branch-free inner loop (clamped indices, 32-bit B offsets for SADDR
// addressing); epilogue has a wave-uniform full-tile fast path.
// mode: 0 plain, 1 +bias then ReLU, 2 +bias, 3 -> 2*acc - I(row==col)
// =====================================================================
__global__ __launch_bounds__(32)
void gemm_wmma(const float* __restrict__ Ag, const float* __restrict__ Bg,
               const float* __restrict__ bias, float* __restrict__ Cg,
               int M, int N, int K, int lda, int ldb, int ldc,
               long long sA, long long sB, long long sC, int mode)
{
  const float* A = Ag + (size_t)blockIdx.z * (size_t)sA;
  const float* B = Bg + (size_t)blockIdx.z * (size_t)sB;
  float*       C = Cg + (size_t)blockIdx.z * (size_t)sC;
  const int lane = threadIdx.x;
  const int mr   = lane & 15;
  const int half = lane >> 4;
  const int row0 = blockIdx.y << 6;         // 64-row tile
  const int col0 = blockIdx.x << 5;         // 32-col tile

  // clamped load indices (branch-free; junk lanes only feed junk outputs)
  const float* Arp[4];
#pragma unroll
  for (int mi = 0; mi < 4; ++mi) {
    int ar = min(row0 + 16*mi + mr, M - 1);
    Arp[mi] = A + (size_t)ar * lda + 2*half;      // frag elems contiguous
  }
  const int bc0 = min(col0 + mr,      N - 1);
  const int bc1 = min(col0 + 16 + mr, N - 1);

  v8f acc[8];
#pragma unroll
  for (int i = 0; i < 8; ++i) acc[i] = (v8f){0.f,0.f,0.f,0.f,0.f,0.f,0.f,0.f};

  // 32-bit offsets -> uniform-base + vgpr-offset addressing for B
  unsigned boff0 = (unsigned)(2*half) * (unsigned)ldb + (unsigned)bc0;
  unsigned boff1 = (unsigned)(2*half) * (unsigned)ldb + (unsigned)bc1;
  const unsigned bstep = 4u * (unsigned)ldb;

  for (int k0 = 0; k0 < K; k0 += 4) {
    v2f a[4];
#pragma unroll
    for (int mi = 0; mi < 4; ++mi)
      a[mi] = *reinterpret_cast<const v2f*>(Arp[mi] + k0);   // global_load_b64
    v2f b0, b1;
    b0[0] = B[boff0]; b0[1] = B[boff0 + (unsigned)ldb];
    b1[0] = B[boff1]; b1[1] = B[boff1 + (unsigned)ldb];
    boff0 += bstep; boff1 += bstep;
#pragma unroll
    for (int mi = 0; mi < 4; ++mi) {
      acc[mi*2 + 0] = WMMA_F32(a[mi], b0, acc[mi*2 + 0]);
      acc[mi*2 + 1] = WMMA_F32(a[mi], b1, acc[mi*2 + 1]);
    }
  }

  const bool full = (row0 + 63 < M) && (col0 + 31 < N);   // wave-uniform
  if (full) {
#pragma unroll
    for (int mi = 0; mi < 4; ++mi) {
#pragma unroll
      for (int ni = 0; ni < 2; ++ni) {
        const int colb = col0 + ni*16 + mr;
        const v8f& av = acc[mi*2 + ni];
#pragma unroll
        for (int v = 0; v < 8; ++v) {
          const int row = row0 + mi*16 + v + 8*half;
          float val = av[v];
          if      (mode == 1) val = fmaxf(val + bias[colb], 0.f);
          else if (mode == 2) val = val + bias[colb];
          else if (mode == 3) val = 2.f*val - ((row == colb) ? 1.f : 0.f);
          C[(size_t)row * ldc + colb] = val;
        }
      }
    }
  } else {
#pragma unroll
    for (int mi = 0; mi < 4; ++mi) {
#pragma unroll
      for (int ni = 0; ni < 2; ++ni) {
        const int colb = col0 + ni*16 + mr;
        const v8f& av = acc[mi*2 + ni];
#pragma unroll
        for (int v = 0; v < 8; ++v) {
          const int row = row0 + mi*16 + v + 8*half;
          if (row < M && colb < N) {
            float val = av[v];
            if      (mode == 1) val = fmaxf(val + bias[colb], 0.f);
            else if (mode == 2) val = val + bias[colb];
            else if (mode == 3) val = 2.f*val - ((row == colb) ? 1.f : 0.f);
            C[(size_t)row * ldc + colb] = val;
          }
        }
      }
    }
  }
}

// =====================================================================
// Depthwise causal conv (K=4) + bias + SiLU over xz[:, :DI]
// =====================================================================
__global__ void conv_silu(const float* __restrict__ xz, const float* __restrict__ cw,
                          const float* __restrict__ cb, float* __restrict__ xc, int total)
{
  int i = blockIdx.x * blockDim.x + threadIdx.x;
  if (i >= total) return;
  int d  = i & (DI - 1);
  int bt = i >> 7;            // row = b*L + t
  int t  = bt & (LL - 1);
  int b  = bt >> 10;
  float acc = cb[d];
#pragma unroll
  for (int k = 0; k < KK; ++k) {
    int tp = t - (KK - 1) + k;
    if (tp >= 0)
      acc += xz[((size_t)(b * LL + tp)) * (2*DI) + d] * cw[d * KK + k];
  }
  xc[i] = acc / (1.f + __expf(-acc));      // silu
}

// =====================================================================
// Fused selective-scan: dt projection+softplus, recurrence, C-contract,
// skip D*x and SiLU(z) gate.  One block per batch; 2 threads per channel,
// 8 states each; cross-thread reduce via wave32 shfl_xor.
// =====================================================================
__global__ __launch_bounds__(256)
void mamba_scan(const float* __restrict__ xz, const float* __restrict__ xc,
                const float* __restrict__ dbc, const float* __restrict__ Wdt,
                const float* __restrict__ bdt, const float* __restrict__ Alog,
                const float* __restrict__ Dp,  float* __restrict__ y)
{
  const int b   = blockIdx.x;
  const int tid = threadIdx.x;
  const int d   = tid >> 1;
  const int sh  = (tid & 1) * 8;

  __shared__ float sB[SS], sC[SS], sdt4[RR];

  float A[8], h[8];
#pragma unroll
  for (int j = 0; j < 8; ++j) { A[j] = -__expf(Alog[d * SS + sh + j]); h[j] = 0.f; }
  const float w0 = Wdt[0*DI + d], w1 = Wdt[1*DI + d], w2 = Wdt[2*DI + d], w3 = Wdt[3*DI + d];
  const float bd = bdt[d], dpd = Dp[d];

  for (int t = 0; t < LL; ++t) {
    size_t row = (size_t)b * LL + t;
    __syncthreads();
    if (tid < 16)       sB[tid]        = dbc[row * 36 + RR + tid];
    else if (tid < 32)  sC[tid - 16]   = dbc[row * 36 + RR + SS + (tid - 16)];
    else if (tid < 36)  sdt4[tid - 32] = dbc[row * 36 + (tid - 32)];
    __syncthreads();

    float v   = sdt4[0]*w0 + sdt4[1]*w1 + sdt4[2]*w2 + sdt4[3]*w3 + bd;
    float dtv = (v > 20.f) ? v : log1pf(__expf(v));      // softplus
    float xcv = xc[row * DI + d];
    float acc = 0.f;
#pragma unroll
    for (int j = 0; j < 8; ++j) {
      h[j] = h[j] * __expf(dtv * A[j]) + dtv * sB[sh + j] * xcv;
      acc += h[j] * sC[sh + j];
    }
    acc += __shfl_xor(acc, 1, 32);       // combine the two half-threads of this channel
    if ((tid & 1) == 0) {
      float zv  = xz[row * (2*DI) + DI + d];
      float gat = zv / (1.f + __expf(-zv));
      y[row * DI + d] = (acc + dpd * xcv) * gat;
    }
  }
}

// =====================================================================
// Flip along sequence dim: dst[b,t] = src[b,L-1-t]
// =====================================================================
__global__ void flip_seq(const float* __restrict__ src, float* __restrict__ dst, int total)
{
  int i = blockIdx.x * blockDim.x + threadIdx.x;
  if (i >= total) return;
  int d  = i & (DD - 1);
  int lt = (i >> 6) & (LL - 1);
  int b  = i >> 16;
  dst[i] = src[((size_t)b << 16) + (size_t)(LL - 1 - lt) * DD + d];
}

// =====================================================================
// out = LayerNorm(a + b? + c?(optionally time-flipped)) ; one thread/row
// =====================================================================
__global__ void add_ln(const float* __restrict__ a, const float* __restrict__ bsrc,
                       const float* __restrict__ csrc, int flipC,
                       const float* __restrict__ g, const float* __restrict__ beta,
                       float* __restrict__ out, int rows)
{
  int r = blockIdx.x * blockDim.x + threadIdx.x;
  if (r >= rows) return;
  int b = r >> 10, t = r & (LL - 1);
  size_t ra = (size_t)r * DD;
  size_t rc = flipC ? ((size_t)(b * LL + (LL - 1 - t))) * DD : ra;
  float v[DD];
  float m = 0.f;
#pragma unroll 4
  for (int j = 0; j < DD; ++j) {
    float x = a[ra + j];
    if (bsrc) x += bsrc[ra + j];
    if (csrc) x += csrc[rc + j];
    v[j] = x; m += x;
  }
  m *= (1.f / DD);
  float var = 0.f;
#pragma unroll 4
  for (int j = 0; j < DD; ++j) { float dl = v[j] - m; var += dl * dl; }
  var *= (1.f / DD);
  float inv = rsqrtf(var + 1e-5f);
#pragma unroll 4
  for (int j = 0; j < DD; ++j)
    out[ra + j] = (v[j] - m) * inv * g[j] + beta[j];
}

// =====================================================================
// sup = softmax(relu(E @ E^T)) row-wise.  One block per row.
// =====================================================================
__global__ __launch_bounds__(256)
void sup_softmax(const float* __restrict__ E, float* __restrict__ sup)
{
  const int n = blockIdx.x, tid = threadIdx.x;
  __shared__ float er[EDIM];
  __shared__ float red[256];
  if (tid < EDIM) er[tid] = E[n * EDIM + tid];
  __syncthreads();

  float s[4]; float mx = 0.f;           // relu => scores >= 0
#pragma unroll
  for (int q = 0; q < 4; ++q) {
    int m = tid + q * 256;
    float dot = 0.f;
#pragma unroll
    for (int e = 0; e < EDIM; ++e) dot += er[e] * E[m * EDIM + e];
    dot = fmaxf(dot, 0.f);
    s[q] = dot; mx = fmaxf(mx, dot);
  }
  red[tid] = mx; __syncthreads();
  for (int st = 128; st > 0; st >>= 1) { if (tid < st) red[tid] = fmaxf(red[tid], red[tid + st]); __syncthreads(); }
  mx = red[0]; __syncthreads();

  float sum = 0.f;
#pragma unroll
  for (int q = 0; q < 4; ++q) { s[q] = __expf(s[q] - mx); sum += s[q]; }
  red[tid] = sum; __syncthreads();
  for (int st = 128; st > 0; st >>= 1) { if (tid < st) red[tid] += red[tid + st]; __syncthreads(); }
  float inv = 1.f / red[0];
#pragma unroll
  for (int q = 0; q < 4; ++q) sup[(size_t)n * NN + tid + q * 256] = s[q] * inv;
}

// =====================================================================
// Per-node adaptive pooling:  out[b,n,o] = sum_{k,c} xg_k[b,n,c]*W_n[k,c,o] + bias_n[o]
// W_n built in LDS from node_emb @ W_pool; GEMM 16x192x64 via WMMA (4 waves).
// =====================================================================
__global__ __launch_bounds__(128)
void node_pool(const float* __restrict__ xg0, const float* __restrict__ xg1,
               const float* __restrict__ xg2, const float* __restrict__ E,
               const float* __restrict__ Wp,  const float* __restrict__ bp,
               float* __restrict__ outp)
{
  const int n = blockIdx.x, tid = threadIdx.x;
  __shared__ float Wn[3 * DD * DD];      // 192 x 64, row-major over (k*64+c, o)
  __shared__ float bn[DD];

  float e[EDIM];
#pragma unroll
  for (int q = 0; q < EDIM; ++q) e[q] = E[n * EDIM + q];

  for (int j = tid; j < 3 * DD * DD; j += 128) {
    float s = 0.f;
#pragma unroll
    for (int q = 0; q < EDIM; ++q) s += e[q] * Wp[(size_t)q * (3 * DD * DD) + j];
    Wn[j] = s;
  }
  if (tid < DD) {
    float s = 0.f;
#pragma unroll
    for (int q = 0; q < EDIM; ++q) s += e[q] * bp[q * DD + tid];
    bn[tid] = s;
  }
  __syncthreads();

  const int lane = tid & 31, wv = tid >> 5;
  const int mr = lane & 15, half = lane >> 4;
  v8f acc = {0.f,0.f,0.f,0.f,0.f,0.f,0.f,0.f};
  for (int k0 = 0; k0 < 192; k0 += 4) {
    v2f a, b;
#pragma unroll
    for (int v = 0; v < 2; ++v) {
      int kk = k0 + v + 2*half;
      int ks = kk >> 6, c = kk & 63;
      const float* p = (ks == 0) ? xg0 : ((ks == 1) ? xg1 : xg2);
      a[v] = p[(size_t)mr * (NN * DD) + (size_t)n * DD + c];
      b[v] = Wn[kk * DD + wv * 16 + mr];
    }
    acc = WMMA_F32(a, b, acc);
  }
#pragma unroll
  for (int v = 0; v < 8; ++v) {
    int bb = v + 8*half;
    int o  = wv * 16 + mr;
    outp[(size_t)bb * (NN * DD) + (size_t)n * DD + o] = acc[v] + bn[o];
  }
}

// =====================================================================
// Host-side helpers
// =====================================================================
static void launch_gemm(const float* A, const float* B, const float* bias, float* C,
                        int M, int N, int K, int lda, int ldb, int ldc,
                        long long sA, long long sB, long long sC, int batch, int mode,
                        hipStream_t s)
{
  dim3 grid((N + 31) / 32, (M + 63) / 64, batch);
  gemm_wmma<<<grid, dim3(32), 0, s>>>(A, B, bias, C, M, N, K, lda, ldb, ldc, sA, sB, sC, mode);
}

static void run_dir(const float* xin, const float* const* w, float* ws, float* yout,
                    hipStream_t s)
{
  // w: [0]=Wi [1]=conv_w [2]=conv_b [3]=Wx [4]=Wdt [5]=dtb [6]=Alog [7]=D [8]=Wo
  float* xz  = ws + O_XZ;
  float* xc  = ws + O_XC;
  float* dbc = ws + O_DBC;
  float* yg  = ws + O_YBUF;
  const int rows = BB * LL;                       // 16384
  launch_gemm(xin, w[0], nullptr, xz, rows, 2*DI, DD, DD, 2*DI, 2*DI, 0,0,0, 1, 0, s);
  conv_silu<<<(rows * DI + 255) / 256, 256, 0, s>>>(xz, w[1], w[2], xc, rows * DI);
  launch_gemm(xc, w[3], nullptr, dbc, rows, RR + 2*SS, DI, DI, RR + 2*SS, RR + 2*SS, 0,0,0, 1, 0, s);
  mamba_scan<<<BB, 256, 0, s>>>(xz, xc, dbc, w[4], w[5], w[6], w[7], yg);
  launch_gemm(yg, w[8], nullptr, yout, rows, DD, DI, DI, DD, DD, 0,0,0, 1, 0, s);
}

extern "C" void kernel_launch(void* const* d_in, const int* in_sizes, int n_in,
                              void* d_out, int out_size, void* d_ws, size_t ws_size,
                              hipStream_t stream)
{
  (void)in_sizes; (void)n_in; (void)out_size; (void)ws_size;
  const float* x_in = (const float*)d_in[0];
  const float* fw[9]; const float* bw[9];
  for (int i = 0; i < 9; ++i) { fw[i] = (const float*)d_in[1 + i]; bw[i] = (const float*)d_in[10 + i]; }
  const float* ln1g = (const float*)d_in[19];
  const float* ln1b = (const float*)d_in[20];
  const float* w1   = (const float*)d_in[21];
  const float* b1   = (const float*)d_in[22];
  const float* w2   = (const float*)d_in[23];
  const float* b2   = (const float*)d_in[24];
  const float* ln2g = (const float*)d_in[25];
  const float* ln2b = (const float*)d_in[26];
  const float* E    = (const float*)d_in[27];
  const float* Wp   = (const float*)d_in[28];
  const float* bp   = (const float*)d_in[29];
  const float* pw   = (const float*)d_in[30];
  const float* pb   = (const float*)d_in[31];
  float* ws = (float*)d_ws;
  const int rows = BB * LL;

  for (int layer = 0; layer < 2; ++layer) {
    const float* xcur = layer ? (ws + O_X) : x_in;
    // forward direction
    run_dir(xcur, fw, ws, ws + O_YF, stream);
    // backward direction on flipped sequence (result stays flipped; add_ln re-flips)
    flip_seq<<<(rows * DD + 255) / 256, 256, 0, stream>>>(xcur, ws + O_XFLIP, rows * DD);
    run_dir(ws + O_XFLIP, bw, ws, ws + O_YB, stream);
    // x = LN(x + yf + flip(yb))
    add_ln<<<(rows + 255) / 256, 256, 0, stream>>>(xcur, ws + O_YF, ws + O_YB, 1,
                                                   ln1g, ln1b, ws + O_X, rows);
    // FFN: h = relu(x@W1+b1); x = LN(x + h@W2 + b2)
    launch_gemm(ws + O_X, w1, b1, ws + O_XZ, rows, HH, DD, DD, HH, HH, 0,0,0, 1, 1, stream);
    launch_gemm(ws + O_XZ, w2, b2, ws + O_XC, rows, DD, HH, HH, DD, DD, 0,0,0, 1, 2, stream);
    add_ln<<<(rows + 255) / 256, 256, 0, stream>>>(ws + O_X, ws + O_XC, nullptr, 0,
                                                   ln2g, ln2b, ws + O_X, rows);
  }

  // Graph stage
  sup_softmax<<<NN, 256, 0, stream>>>(E, ws + O_SUP);
  // cheb2 = 2*sup@sup - I
  launch_gemm(ws + O_SUP, ws + O_SUP, nullptr, ws + O_CHEB2,
              NN, NN, NN, NN, NN, NN, 0,0,0, 1, 3, stream);
  // xg1[b] = sup @ x_b ; xg2[b] = cheb2 @ x_b  (batched over b)
  launch_gemm(ws + O_SUP,   ws + O_X, nullptr, ws + O_XG1,
              NN, DD, NN, NN, DD, DD, 0, (long long)NN*DD, (long long)NN*DD, BB, 0, stream);
  launch_gemm(ws + O_CHEB2, ws + O_X, nullptr, ws + O_XG2,
              NN, DD, NN, NN, DD, DD, 0, (long long)NN*DD, (long long)NN*DD, BB, 0, stream);
  // per-node pooling (+node bias)
  node_pool<<<NN, 128, 0, stream>>>(ws + O_X, ws + O_XG1, ws + O_XG2, E, Wp, bp, ws + O_OUTPRE);
  // final projection
  launch_gemm(ws + O_OUTPRE, pw, pb, (float*)d_out, rows, PLDIM, DD, DD, PLDIM, PLDIM,
              0,0,0, 1, 2, stream);
}